// Net_56573309224519
// MI455X (gfx1250) — compile-verified
//
#include <hip/hip_runtime.h>
#include <hip/hip_bf16.h>

// Problem constants (from reference)
#define BB 8192
#define LL 100
#define DD 128
#define NLAB 20
#define NPAD 32     // labels padded to 32 (two 16-wide WMMA N-tiles)

typedef __attribute__((ext_vector_type(2))) float v2f;
typedef __attribute__((ext_vector_type(8))) float v8f;

// ---------------------------------------------------------------------------
// Kernel 0: pad W[20][128] -> Wpad[32][128] (zero rows 20..31) and
// bias[20] -> biaspad[32]. Removes all divergence from the WMMA hot loop.
// ---------------------------------------------------------------------------
__global__ __launch_bounds__(128) void pad_weights_kernel(
    const float* __restrict__ W,
    const float* __restrict__ bias,
    float* __restrict__ Wpad,
    float* __restrict__ biaspad)
{
    const int tid = threadIdx.x;                 // 0..127 = k index
    #pragma unroll
    for (int n = 0; n < NPAD; ++n)
        Wpad[n * DD + tid] = (n < NLAB) ? W[n * DD + tid] : 0.0f;
    if (tid < NPAD)
        biaspad[tid] = (tid < NLAB) ? bias[tid] : 0.0f;
}

// ---------------------------------------------------------------------------
// Kernel 1: fused gather + masked mean.
// One wave32 per sentence; each lane owns a float4 slice of the 128-float
// embedding row (16B/lane, fully coalesced 512B per row). Tokens staged in
// LDS; next row prefetched (global_prefetch_b8) to hide the dependent gather
// latency — the 51.2MB table is L2-resident on MI455X (192MB L2).
// ---------------------------------------------------------------------------
__global__ __launch_bounds__(128) void embbag_mean_kernel(
    const int* __restrict__ tokens,
    const int* __restrict__ lengths,
    const float4* __restrict__ emb4,   // [V][32] float4
    float4* __restrict__ x4)           // [B][32] float4
{
    const int wave = threadIdx.x >> 5;           // 0..3
    const int lane = threadIdx.x & 31;
    const int b    = blockIdx.x * 4 + wave;

    __shared__ int toks[4][LL + 4];

    const int len = lengths[b];
    for (int l = lane; l < len; l += 32)
        toks[wave][l] = tokens[b * LL + l];
    __syncthreads();

    float4 acc = make_float4(0.f, 0.f, 0.f, 0.f);
    for (int l = 0; l < len; ++l) {
        const int t = toks[wave][l];
        if (l + 1 < len) {
            const int tn = toks[wave][l + 1];
            __builtin_prefetch(&emb4[(size_t)tn * 32 + lane], 0, 3);
        }
        const float4 e = emb4[(size_t)t * 32 + lane];
        acc.x += e.x; acc.y += e.y; acc.z += e.z; acc.w += e.w;
    }

    const float inv = 1.0f / (float)len;
    acc.x *= inv; acc.y *= inv; acc.z *= inv; acc.w *= inv;
    x4[(size_t)b * 32 + lane] = acc;
}

// ---------------------------------------------------------------------------
// Kernel 2: out[B,20] = x[B,128] @ Wpad[32,128]^T + biaspad, full f32 via
// V_WMMA_F32_16X16X4_F32. Each wave owns one 16-row M-tile and both 16-col
// N-tiles; K=128 walked in 32 steps of 4. All operand loads unconditional
// (Wpad pre-zeroed) -> inner loop is 3x global_load_b64 + 2x wmma.
//
// Fragment layouts (ISA 7.12.2, wave32):
//   A 16x4 f32 : lane = 16*khalf + m ; V0 = A[m][2*khalf], V1 = A[m][2*khalf+1]
//   B  4x16 f32: lane = 16*khalf + n ; V0 = B[2*khalf][n], V1 = B[2*khalf+1][n]
//   C/D 16x16  : VGPR r: row = r + 8*(lane>=16), col = lane%16
// ---------------------------------------------------------------------------
__global__ __launch_bounds__(256) void head_wmma_kernel(
    const float* __restrict__ x,        // [B][128]
    const float* __restrict__ Wpad,     // [32][128], rows 20..31 zero
    const float* __restrict__ biaspad,  // [32], entries 20..31 zero
    float* __restrict__ out)            // [B][20]
{
    const int wave  = threadIdx.x >> 5;              // 0..7
    const int lane  = threadIdx.x & 31;
    const int tile  = blockIdx.x * 8 + wave;         // 0..511
    const int m0    = tile * 16;

    const int mlo   = lane & 15;                     // M (for A) / N (for B)
    const int khalf = lane >> 4;                     // 0 or 1
    const int kofs  = khalf * 2;

    const float* xrow = x    + (size_t)(m0 + mlo) * DD + kofs;
    const float* wb0  = Wpad + (size_t)mlo * DD        + kofs;   // N-tile 0
    const float* wb1  = wb0  + 16 * DD;                          // N-tile 1

    v8f c0 = {};
    v8f c1 = {};

    #pragma unroll 4
    for (int ks = 0; ks < 32; ++ks) {
        const int kb = ks * 4;

        v2f a, b0, b1;
        a.x  = xrow[kb];  a.y  = xrow[kb + 1];
        b0.x = wb0[kb];   b0.y = wb0[kb + 1];
        b1.x = wb1[kb];   b1.y = wb1[kb + 1];

        c0 = __builtin_amdgcn_wmma_f32_16x16x4_f32(
                 false, a, false, b0, (short)0, c0, false, false);
        c1 = __builtin_amdgcn_wmma_f32_16x16x4_f32(
                 false, a, false, b1, (short)0, c1, false, false);
    }

    // Epilogue: add bias, scatter the 16x20 valid region of the 16x32 tile.
    const int col    = lane & 15;
    const int rowOff = (lane >> 4) * 8;
    const float bias0 = biaspad[col];
    const float bias1 = biaspad[col + 16];

    #pragma unroll
    for (int r = 0; r < 8; ++r) {
        const int row = m0 + rowOff + r;
        out[(size_t)row * NLAB + col] = c0[r] + bias0;
        if (col + 16 < NLAB)
            out[(size_t)row * NLAB + col + 16] = c1[r] + bias1;
    }
}

extern "C" void kernel_launch(void* const* d_in, const int* in_sizes, int n_in,
                              void* d_out, int out_size, void* d_ws, size_t ws_size,
                              hipStream_t stream) {
    const int*   tokens  = (const int*)  d_in[0];   // [B, L] int32
    const int*   lengths = (const int*)  d_in[1];   // [B]    int32
    const float* emb     = (const float*)d_in[2];   // [V, D] f32
    const float* W       = (const float*)d_in[3];   // [20, D] f32
    const float* bias    = (const float*)d_in[4];   // [20]   f32
    float*       out     = (float*)d_out;           // [B, 20] f32

    // Workspace layout: x[B*D] | Wpad[32*D] | biaspad[32]
    float* x       = (float*)d_ws;                  // 4 MB
    float* Wpad    = x + (size_t)BB * DD;           // 16 KB
    float* biaspad = Wpad + NPAD * DD;              // 128 B

    pad_weights_kernel<<<1, 128, 0, stream>>>(W, bias, Wpad, biaspad);

    // Kernel 1: 4 sentences per 128-thread block (1 wave each).
    embbag_mean_kernel<<<BB / 4, 128, 0, stream>>>(
        tokens, lengths, (const float4*)emb, (float4*)x);

    // Kernel 2: 8 waves per block, one 16-row tile per wave -> 64 blocks.
    head_wmma_kernel<<<BB / 16 / 8, 256, 0, stream>>>(x, Wpad, biaspad, out);
}